// S4_40673340293925
// MI455X (gfx1250) — compile-verified
//
#include <hip/hip_runtime.h>

// S4 (HiPPO-LegS, bilinear-discretized) via exact chunked-scan reformulation.
// Matrix work: V_WMMA_F32_16X16X4_F32 (fp32 end-to-end; a 4096-deep recurrence
// must not run in bf16). Data movement: gfx1250 async direct-to-LDS loads
// (GLOBAL_LOAD_ASYNC_TO_LDS_B32/B128, ASYNCcnt) for all LDS staging.
// Workspace requirement: ~36 MB.

typedef float v2f __attribute__((ext_vector_type(2)));
typedef float v8f __attribute__((ext_vector_type(8)));

#define BATCH 16
#define SEQ   4096
#define DIN   64
#define NH    64
#define LCH   64            // chunk length
#define NCH   (SEQ / LCH)   // 64 chunks
#define LD    68            // padded LDS leading dim: 272B rows -> 16B aligned,
                            // +4 bank shift per row -> conflict-free col gathers

// workspace offsets (in floats)
#define OFF_BM 0u            // Bm[d][n][j]      : 64*64*64
#define OFF_G  262144u       // G [d][j][n]      : 64*64*64
#define OFF_K  524288u       // k [d][j]         : 64*64
#define OFF_AL 528384u       // dA^64            : 64*64
#define OFF_ST 532480u       // states[c][d][n][b]: 64*64*64*16
#define OFF_YM 4726784u      // Ymid[b][t][d]    : 16*4096*64
// total = 8,921,088 floats = 35.7 MB

// ---- gfx1250 async direct-to-LDS copies (CDNA5 ISA) ------
__device__ __forceinline__ void async_b32(void* lds, const void* gaddr) {
  unsigned l = (unsigned)(size_t)lds;                 // LDS byte offset
  unsigned long long g = (unsigned long long)(size_t)gaddr;
  asm volatile("global_load_async_to_lds_b32 %0, %1, off"
               :: "v"(l), "v"(g) : "memory");
}
__device__ __forceinline__ void async_b128(void* lds, const void* gaddr) {
  unsigned l = (unsigned)(size_t)lds;
  unsigned long long g = (unsigned long long)(size_t)gaddr;
  asm volatile("global_load_async_to_lds_b128 %0, %1, off"
               :: "v"(l), "v"(g) : "memory");
}
__device__ __forceinline__ void wait_async0() {
  asm volatile("s_wait_asynccnt 0x0" ::: "memory");
}

// ---- 64x64x64 fp32 GEMM on LDS tiles, 8 waves (blockDim.x == 256) --------
// Wave w owns row-tile mt = w>>1 and col-tiles ntBase, ntBase+1 so the
// A-fragment is loaded once and reused by two WMMAs per K-step.
__device__ inline void gemm64_8w(const float* A, const float* B, float* C) {
  const int wave = threadIdx.x >> 5;
  const int lane = threadIdx.x & 31;
  const int l    = lane & 15;
  const int hi   = lane >> 4;
  const int mt   = wave >> 1;
  const int nt0  = (wave & 1) * 2;
  v8f acc0 = {}, acc1 = {};
#pragma unroll
  for (int k0 = 0; k0 < 64; k0 += 4) {
    v2f a, b0, b1;
    a.x  = A[(mt * 16 + l) * LD + k0 + 2 * hi];
    a.y  = A[(mt * 16 + l) * LD + k0 + 2 * hi + 1];
    b0.x = B[(k0 + 2 * hi) * LD + nt0 * 16 + l];
    b0.y = B[(k0 + 2 * hi + 1) * LD + nt0 * 16 + l];
    b1.x = B[(k0 + 2 * hi) * LD + (nt0 + 1) * 16 + l];
    b1.y = B[(k0 + 2 * hi + 1) * LD + (nt0 + 1) * 16 + l];
    acc0 = __builtin_amdgcn_wmma_f32_16x16x4_f32(false, a, false, b0,
                                                 (short)0, acc0, false, false);
    acc1 = __builtin_amdgcn_wmma_f32_16x16x4_f32(false, a, false, b1,
                                                 (short)0, acc1, false, false);
  }
#pragma unroll
  for (int v = 0; v < 8; ++v) {
    C[(mt * 16 + hi * 8 + v) * LD + nt0 * 16 + l]       = acc0[v];
    C[(mt * 16 + hi * 8 + v) * LD + (nt0 + 1) * 16 + l] = acc1[v];
  }
}

// ---- Kernel 1: precompute Bm, G, k, dA^64 (3 independent blocks) ---------
__global__ __launch_bounds__(256) void s4_precompute(
    const float* __restrict__ dA, const float* __restrict__ dB,
    const float* __restrict__ Cp, float* __restrict__ ws) {
  __shared__ __align__(16) float sm[4 * 64 * LD];
  float* b0 = sm;
  float* b1 = sm + 64 * LD;
  float* b2 = sm + 2 * 64 * LD;
  float* b3 = sm + 3 * 64 * LD;
  const int tid = threadIdx.x;
  float* Bm = ws + OFF_BM;
  float* G  = ws + OFF_G;
  float* kk = ws + OFF_K;
  float* AL = ws + OFF_AL;

  if (blockIdx.x == 0) {
    // chain W_s = dA^s @ DB (DB[n][d] = dB[d][n]);  Bm[d][n][63-s] = W_s[n][d]
    for (int g = tid; g < 1024; g += 256) {   // dA rows, async B128
      int row = g >> 4, col = (g & 15) * 4;
      async_b128(&b0[row * LD + col], dA + g * 4);
    }
    for (int i = tid; i < 4096; i += 256)     // transposed scatter: plain
      b1[(i & 63) * LD + (i >> 6)] = dB[i];   // dB (D,N) -> (N,D)
    wait_async0();
    __syncthreads();
    float* cur = b1; float* nxt = b2;
    for (int s = 0; s < 64; ++s) {
      for (int i = tid; i < 4096; i += 256) {
        int n = i >> 6, d = i & 63;
        Bm[d * 4096 + n * 64 + (63 - s)] = cur[n * LD + d];
      }
      if (s < 63) {
        gemm64_8w(b0, cur, nxt);
        __syncthreads();
        float* tmp = cur; cur = nxt; nxt = tmp;
      }
    }
  } else if (blockIdx.x == 1) {
    // chain g_j = Cp @ dA^{j+1};  G[d][j][n] = g_j[d][n];  k[d][j] taps
    for (int g = tid; g < 1024; g += 256) {
      int row = g >> 4, col = (g & 15) * 4;
      async_b128(&b0[row * LD + col], dA + g * 4);
      async_b128(&b3[row * LD + col], dB + g * 4);
      async_b128(&b1[row * LD + col], Cp + g * 4);
    }
    wait_async0();
    __syncthreads();
    if (tid < 64) {  // k[d][0] = C_d . dB_d
      float s = 0.f;
      for (int n = 0; n < 64; ++n) s += b1[tid * LD + n] * b3[tid * LD + n];
      kk[tid * 64 + 0] = s;
    }
    gemm64_8w(b1, b0, b2);  // g_0 = Cp @ dA
    __syncthreads();
    float* cur = b2; float* nxt = b1;
    for (int j = 0; j < 64; ++j) {
      for (int i = tid; i < 4096; i += 256) {
        int d = i >> 6, n = i & 63;
        G[d * 4096 + j * 64 + n] = cur[d * LD + n];
      }
      if (j < 63) {
        if (tid < 64) {  // k[d][j+1] = g_j[d] . dB_d
          float s = 0.f;
          for (int n = 0; n < 64; ++n) s += cur[tid * LD + n] * b3[tid * LD + n];
          kk[tid * 64 + (j + 1)] = s;
        }
        gemm64_8w(cur, b0, nxt);
        __syncthreads();
        float* tmp = cur; cur = nxt; nxt = tmp;
      }
    }
  } else {
    // AL = dA^64 by six squarings
    for (int g = tid; g < 1024; g += 256) {
      int row = g >> 4, col = (g & 15) * 4;
      async_b128(&b0[row * LD + col], dA + g * 4);
    }
    wait_async0();
    __syncthreads();
    float* cur = b0; float* nxt = b1;
    for (int it = 0; it < 6; ++it) {
      gemm64_8w(cur, cur, nxt);
      __syncthreads();
      float* tmp = cur; cur = nxt; nxt = tmp;
    }
    for (int i = tid; i < 4096; i += 256)
      AL[i] = cur[(i >> 6) * LD + (i & 63)];
  }
}

// ---- Kernel 2: chunk-boundary state scan (one block per channel d) -------
// S_{c+1}(NxB) = AL @ S_c + Bm_d @ U_c ; store S_c for all chunks.
__global__ __launch_bounds__(128) void s4_scan(
    const float* __restrict__ x, float* __restrict__ ws) {
  const int d    = blockIdx.x;
  const int tid  = threadIdx.x;
  const int wave = tid >> 5, lane = tid & 31, l = lane & 15, hi = lane >> 4;
  __shared__ __align__(16) float ALs[64 * LD];
  __shared__ __align__(16) float Bms[64 * LD];
  __shared__ __align__(16) float S0[64 * 16];
  __shared__ __align__(16) float S1[64 * 16];
  __shared__ __align__(16) float U[64 * 16];
  const float* ALg = ws + OFF_AL;
  const float* Bmg = ws + OFF_BM + d * 4096;
  float* st = ws + OFF_ST;

  for (int g = tid; g < 1024; g += 128) {
    int row = g >> 4, col = (g & 15) * 4;
    async_b128(&ALs[row * LD + col], ALg + g * 4);
    async_b128(&Bms[row * LD + col], Bmg + g * 4);
  }
  for (int i = tid; i < 1024; i += 128) {
    S0[i] = 0.f;
    st[(0 * 64 + d) * 1024 + i] = 0.f;  // S_0 = 0
  }
  wait_async0();
  __syncthreads();

  float* Sc = S0; float* Sn = S1;
  for (int c = 0; c < NCH; ++c) {
    // U[j][b] = x[b][c*64+j][d] : stride-64 gather, async b32 (L2-resident)
    for (int i = tid; i < 1024; i += 128) {
      int j = i >> 4, b = i & 15;
      async_b32(&U[j * 16 + b], &x[((b * SEQ) + c * LCH + j) * DIN + d]);
    }
    wait_async0();
    __syncthreads();
    const int mt = wave;  // 4 waves -> 4 row-tiles, single 16-wide col tile
    v8f acc = {};
#pragma unroll
    for (int k0 = 0; k0 < 64; k0 += 4) {
      v2f a, b;
      a.x = ALs[(mt * 16 + l) * LD + k0 + 2 * hi];
      a.y = ALs[(mt * 16 + l) * LD + k0 + 2 * hi + 1];
      b.x = Sc[(k0 + 2 * hi) * 16 + l];
      b.y = Sc[(k0 + 2 * hi + 1) * 16 + l];
      acc = __builtin_amdgcn_wmma_f32_16x16x4_f32(false, a, false, b,
                                                  (short)0, acc, false, false);
    }
#pragma unroll
    for (int k0 = 0; k0 < 64; k0 += 4) {
      v2f a, b;
      a.x = Bms[(mt * 16 + l) * LD + k0 + 2 * hi];
      a.y = Bms[(mt * 16 + l) * LD + k0 + 2 * hi + 1];
      b.x = U[(k0 + 2 * hi) * 16 + l];
      b.y = U[(k0 + 2 * hi + 1) * 16 + l];
      acc = __builtin_amdgcn_wmma_f32_16x16x4_f32(false, a, false, b,
                                                  (short)0, acc, false, false);
    }
    __syncthreads();
#pragma unroll
    for (int v = 0; v < 8; ++v) {
      int n = mt * 16 + hi * 8 + v;
      Sn[n * 16 + l] = acc[v];
      if (c + 1 < NCH)
        st[((c + 1) * 64 + d) * 1024 + n * 16 + l] = acc[v];
    }
    __syncthreads();
    float* tmp = Sc; Sc = Sn; Sn = tmp;
  }
}

// ---- Kernel 3a: per-(chunk, channel) outputs -----------------------------
// Yg(LxB) = G_d @ S_{c,d} via WMMA; + 64-tap causal conv (VALU) + skip.
__global__ __launch_bounds__(128) void s4_chunk_out(
    const float* __restrict__ x, const float* __restrict__ Dp,
    float* __restrict__ ws) {
  const int c = blockIdx.x, d = blockIdx.y;
  const int tid  = threadIdx.x;
  const int wave = tid >> 5, lane = tid & 31, l = lane & 15, hi = lane >> 4;
  __shared__ __align__(16) float Gs[64 * LD];
  __shared__ __align__(16) float Ss[64 * 16];
  __shared__ __align__(16) float Us[64 * 16];
  __shared__ __align__(16) float Yg[64 * 16];
  __shared__ float ks[64];
  const float* G  = ws + OFF_G + d * 4096;
  const float* kk = ws + OFF_K + d * 64;
  const float* st = ws + OFF_ST + (c * 64 + d) * 1024;
  float* Ym = ws + OFF_YM;

  for (int g = tid; g < 1024; g += 128) {
    int row = g >> 4, col = (g & 15) * 4;
    async_b128(&Gs[row * LD + col], G + g * 4);
  }
  for (int g = tid; g < 256; g += 128)      // Ss: contiguous copy, B128
    async_b128(&Ss[g * 4], st + g * 4);
  for (int i = tid; i < 1024; i += 128) {   // U gather, B32
    int j = i >> 4, b = i & 15;
    async_b32(&Us[j * 16 + b], &x[((b * SEQ) + c * LCH + j) * DIN + d]);
  }
  if (tid < 64) ks[tid] = kk[tid];
  wait_async0();
  __syncthreads();

  const int mt = wave;
  v8f acc = {};
#pragma unroll
  for (int k0 = 0; k0 < 64; k0 += 4) {
    v2f a, b;
    a.x = Gs[(mt * 16 + l) * LD + k0 + 2 * hi];
    a.y = Gs[(mt * 16 + l) * LD + k0 + 2 * hi + 1];
    b.x = Ss[(k0 + 2 * hi) * 16 + l];
    b.y = Ss[(k0 + 2 * hi + 1) * 16 + l];
    acc = __builtin_amdgcn_wmma_f32_16x16x4_f32(false, a, false, b,
                                                (short)0, acc, false, false);
  }
#pragma unroll
  for (int v = 0; v < 8; ++v)
    Yg[(mt * 16 + hi * 8 + v) * 16 + l] = acc[v];
  __syncthreads();

  const float dpd = Dp[d];
  for (int i = tid; i < 1024; i += 128) {
    int j = i >> 4, b = i & 15;
    float y = Yg[j * 16 + b] + dpd * Us[j * 16 + b];
    for (int t2 = 0; t2 <= j; ++t2)
      y += ks[j - t2] * Us[t2 * 16 + b];
    Ym[((b * SEQ) + c * LCH + j) * DIN + d] = y;
  }
}

// ---- Kernel 3b: mixing GEMM out = Ymid @ W^T + bias ----------------------
__global__ __launch_bounds__(256) void s4_mix(
    const float* __restrict__ W, const float* __restrict__ bias,
    const float* __restrict__ ws, float* __restrict__ out) {
  const int r0   = blockIdx.x * 128;
  const int tid  = threadIdx.x;
  const int wave = tid >> 5, lane = tid & 31, l = lane & 15, hi = lane >> 4;
  __shared__ __align__(16) float Ys[128 * LD];
  __shared__ __align__(16) float Ws[64 * LD];
  __shared__ float bs[64];
  const float* Ym = ws + OFF_YM;

  for (int g = tid; g < 2048; g += 256) {   // 128x64 tile, async B128
    int row = g >> 4, col = (g & 15) * 4;
    async_b128(&Ys[row * LD + col], Ym + (size_t)(r0 + row) * 64 + col);
  }
  for (int i = tid; i < 4096; i += 256)     // W[o][d] -> Ws[d][o], async B32
    async_b32(&Ws[(i & 63) * LD + (i >> 6)], W + i);
  if (tid < 64) bs[tid] = bias[tid];
  wait_async0();
  __syncthreads();

  const int mt = wave;  // 8 waves x 16 rows = 128 rows per block
  v8f acc[4] = {{}, {}, {}, {}};
#pragma unroll
  for (int k0 = 0; k0 < 64; k0 += 4) {
    v2f a;
    a.x = Ys[(mt * 16 + l) * LD + k0 + 2 * hi];
    a.y = Ys[(mt * 16 + l) * LD + k0 + 2 * hi + 1];
#pragma unroll
    for (int nt = 0; nt < 4; ++nt) {        // A-frag reused across 4 tiles
      v2f b;
      b.x = Ws[(k0 + 2 * hi) * LD + nt * 16 + l];
      b.y = Ws[(k0 + 2 * hi + 1) * LD + nt * 16 + l];
      acc[nt] = __builtin_amdgcn_wmma_f32_16x16x4_f32(false, a, false, b,
                                                      (short)0, acc[nt],
                                                      false, false);
    }
  }
#pragma unroll
  for (int nt = 0; nt < 4; ++nt) {
    const float bb = bs[nt * 16 + l];
#pragma unroll
    for (int v = 0; v < 8; ++v) {
      int row = r0 + mt * 16 + hi * 8 + v;
      out[(size_t)row * 64 + nt * 16 + l] = acc[nt][v] + bb;
    }
  }
}

extern "C" void kernel_launch(void* const* d_in, const int* in_sizes, int n_in,
                              void* d_out, int out_size, void* d_ws, size_t ws_size,
                              hipStream_t stream) {
  const float* x    = (const float*)d_in[0];
  const float* dA   = (const float*)d_in[1];
  const float* dB   = (const float*)d_in[2];
  const float* Cp   = (const float*)d_in[3];
  const float* Dp   = (const float*)d_in[4];
  const float* W    = (const float*)d_in[5];
  const float* bias = (const float*)d_in[6];
  float* out = (float*)d_out;
  float* ws  = (float*)d_ws;

  s4_precompute<<<3, 256, 0, stream>>>(dA, dB, Cp, ws);
  s4_scan<<<DIN, 128, 0, stream>>>(x, ws);
  s4_chunk_out<<<dim3(NCH, DIN), 128, 0, stream>>>(x, Dp, ws);
  s4_mix<<<(BATCH * SEQ) / 128, 256, 0, stream>>>(W, bias, ws, out);
}